// QuadraticNonLinearity_39213051412534
// MI455X (gfx1250) — compile-verified
//
#include <hip/hip_runtime.h>

// CDNA5 (gfx1250) wave32 WMMA f32 16x16x4 fragments:
//  A (16x4 f32, 2 VGPRs): lane L -> row m = L&15, frag[g] = A[m][g + 2*(L>>4)]
//  B (4x16 f32, 2 VGPRs): lane L -> col n = L&15, frag[g] = B[g + 2*(L>>4)][n]
//  C/D (16x16 f32, 8 VGPRs): lane L, vgpr g -> row m = g + 8*(L>>4), col n = L&15
typedef float v2f __attribute__((ext_vector_type(2)));
typedef float v8f __attribute__((ext_vector_type(8)));

#define NVOX   55296       // 4*24*24*24
#define ABC    13824       // 24*24*24
#define TILES  3456        // NVOX / 16

struct CGPtrs {
  const float* cgr[11];
  const float* cgl[11];
};

// One CG path:  z[v,k,j] += sum_{a,b} y[v,k,a,b] * x[v,j,a,b]
//   x[v,j,a,b] = sum_c rh[l2][v,b,c] * cgr[(j*D1+a)*D2+c]
//   y[v,k,a,b] = sum_c rh[l1][v,c,a] * cgl[(k*D2+b)*D1+c]
// b-slice by b-slice; X_b/Y_b via V_WMMA_F32_16X16X4_F32 batched over the
// 16-voxel tile on N; per-voxel a-contraction on VALU split by (k,j) parity.
// All lane-dependent accesses are branch-free: clamped addresses + cndmask
// on values for loads, cndmask on addresses (dump slot) for stores.
template<int L, int L1, int L2>
__device__ __forceinline__ void do_path(
    const float* __restrict__ cgr,      // (D, D1, D2) row-major, global
    const float* __restrict__ cgl,      // (D, D2, D1) row-major, global
    const float* rh1_lds,               // [16][D1*D1]
    const float* rh2_lds,               // [16][D2*D2]
    float*       out_lds,               // [16][D*D]  (accumulated)
    float*       Xb,                    // [16][MS] scratch
    float*       Yb,                    // [16][MS] scratch
    float*       dump)                  // [32] garbage sink
{
  constexpr int D   = 2*L  + 1;
  constexpr int D1  = 2*L1 + 1;
  constexpr int D2  = 2*L2 + 1;
  constexpr int MS  = D*D1;               // scratch row length
  constexpr int MT  = (MS + 15) / 16;     // row tiles for X
  constexpr int KTX = (D2 + 3) / 4;       // K steps (contract c over D2)
  constexpr int KTY = (D1 + 3) / 4;       // K steps (contract c over D1)
  constexpr int NKJ = (D*D + 1) / 2;      // per-lane z accumulators

  const int lane = threadIdx.x & 31;
  const int half = lane >> 4;             // 0: lanes 0-15, 1: lanes 16-31
  const int lo   = lane & 15;             // voxel-in-tile / row index

  // b-invariant A fragments for X GEMM (rows of cgr), kept in registers
  v2f afx[MT][KTX];
  #pragma unroll
  for (int mt = 0; mt < MT; ++mt)
    #pragma unroll
    for (int kt = 0; kt < KTX; ++kt)
      #pragma unroll
      for (int g = 0; g < 2; ++g) {
        const int m  = mt*16 + lo;
        const int c  = kt*4 + 2*half + g;
        const int mc = m < MS ? m : MS-1;
        const int cc = c < D2 ? c : D2-1;
        const float t = cgr[mc*D2 + cc];                 // unconditional load
        afx[mt][kt][g] = (m < MS && c < D2) ? t : 0.0f;  // v_cndmask
      }

  float zacc[NKJ];
  #pragma unroll
  for (int i = 0; i < NKJ; ++i) zacc[i] = 0.0f;

  for (int b = 0; b < D2; ++b) {
    // ---- X_b[(j*D1+a)] per voxel: B fragments shared across row tiles
    v2f bfx[KTX];
    #pragma unroll
    for (int kt = 0; kt < KTX; ++kt)
      #pragma unroll
      for (int g = 0; g < 2; ++g) {
        const int c  = kt*4 + 2*half + g;
        const int cc = c < D2 ? c : D2-1;
        const float t = rh2_lds[lo*(D2*D2) + b*D2 + cc];
        bfx[kt][g] = (c < D2) ? t : 0.0f;
      }
    #pragma unroll
    for (int mt = 0; mt < MT; ++mt) {
      v8f acc = {0.f,0.f,0.f,0.f,0.f,0.f,0.f,0.f};
      #pragma unroll
      for (int kt = 0; kt < KTX; ++kt)
        acc = __builtin_amdgcn_wmma_f32_16x16x4_f32(
                  false, afx[mt][kt], false, bfx[kt], (short)0, acc, false, false);
      #pragma unroll
      for (int g = 0; g < 8; ++g) {
        const int m = mt*16 + 8*half + g;
        float* p = (m < MS) ? (Xb + lo*MS + m) : (dump + lane);
        *p = acc[g];                                     // unconditional store
      }
    }
    // ---- Y_b[(k*D1+a)] per voxel: A fragments (cgl) hoisted out of a-loop
    v2f afy[KTY];
    #pragma unroll
    for (int kt = 0; kt < KTY; ++kt)
      #pragma unroll
      for (int g = 0; g < 2; ++g) {
        const int k  = lo;
        const int c  = kt*4 + 2*half + g;
        const int kc = k < D  ? k : D-1;
        const int cc = c < D1 ? c : D1-1;
        const float t = cgl[(kc*D2 + b)*D1 + cc];
        afy[kt][g] = (k < D && c < D1) ? t : 0.0f;
      }
    for (int a = 0; a < D1; ++a) {
      v2f bfy[KTY];
      #pragma unroll
      for (int kt = 0; kt < KTY; ++kt)
        #pragma unroll
        for (int g = 0; g < 2; ++g) {
          const int c  = kt*4 + 2*half + g;
          const int cc = c < D1 ? c : D1-1;
          const float t = rh1_lds[lo*(D1*D1) + cc*D1 + a];
          bfy[kt][g] = (c < D1) ? t : 0.0f;
        }
      v8f acc = {0.f,0.f,0.f,0.f,0.f,0.f,0.f,0.f};
      #pragma unroll
      for (int kt = 0; kt < KTY; ++kt)
        acc = __builtin_amdgcn_wmma_f32_16x16x4_f32(
                  false, afy[kt], false, bfy[kt], (short)0, acc, false, false);
      #pragma unroll
      for (int g = 0; g < 8; ++g) {
        const int k = 8*half + g;
        float* p = (k < D) ? (Yb + lo*MS + k*D1 + a) : (dump + lane);
        *p = acc[g];
      }
    }
    __syncthreads();
    // ---- z[v,k,j] += sum_a Y_b[v][k*D1+a] * X_b[v][j*D1+a]   (branch-free)
    #pragma unroll
    for (int i = 0; i < NKJ; ++i) {
      const int kj  = 2*i + half;
      const int kjc = kj < D*D ? kj : D*D-1;
      const int k = kjc / D, j = kjc % D;
      float s = 0.0f;
      #pragma unroll
      for (int a = 0; a < D1; ++a)
        s += Yb[lo*MS + k*D1 + a] * Xb[lo*MS + j*D1 + a];   // contiguous in a
      zacc[i] += (kj < D*D) ? s : 0.0f;
    }
    __syncthreads();
  }
  // flush to LDS accumulator (invalid lanes RMW the dump word)
  #pragma unroll
  for (int i = 0; i < NKJ; ++i) {
    const int kj = 2*i + half;
    float* p = (kj < D*D) ? (out_lds + lo*(D*D) + kj) : (dump + lane);
    *p += zacc[i];
  }
  __syncthreads();
}

__global__ __launch_bounds__(32)
__attribute__((amdgpu_waves_per_eu(4)))
void qnl_kernel(
    const float* __restrict__ rh0, const float* __restrict__ rh2,
    const float* __restrict__ rh4, CGPtrs cg, float* __restrict__ out)
{
  __shared__ __align__(16) float s_in0[16];
  __shared__ __align__(16) float s_in2[16*25];
  __shared__ __align__(16) float s_in4[16*81];
  __shared__ __align__(16) float s_out0[16];
  __shared__ __align__(16) float s_out2[16*25];
  __shared__ __align__(16) float s_out4[16*81];
  __shared__ __align__(16) float s_X[16*81];
  __shared__ __align__(16) float s_Y[16*81];
  __shared__ __align__(16) float s_dump[32];

  const int wg   = blockIdx.x;      // 16-voxel tile
  const int lane = threadIdx.x;

  // vectorized contiguous cooperative loads (16 consecutive voxels)
  {
    const float4* g0 = (const float4*)(rh0 + wg*16);
    const float4* g2 = (const float4*)(rh2 + wg*(16*25));
    const float4* g4 = (const float4*)(rh4 + wg*(16*81));
    float4* l0 = (float4*)s_in0;  float4* z0 = (float4*)s_out0;
    float4* l2 = (float4*)s_in2;  float4* z2 = (float4*)s_out2;
    float4* l4 = (float4*)s_in4;  float4* z4 = (float4*)s_out4;
    const float4 zero = {0.f, 0.f, 0.f, 0.f};
    for (int t = lane; t < 4;   t += 32) { l0[t] = g0[t]; z0[t] = zero; }
    for (int t = lane; t < 100; t += 32) { l2[t] = g2[t]; z2[t] = zero; }
    for (int t = lane; t < 324; t += 32) { l4[t] = g4[t]; z4[t] = zero; }
  }
  __syncthreads();

  // PATHS = [(0,0,0),(0,2,2),(0,4,4),(2,0,2),(2,2,2),(2,2,4),(2,4,4),(4,0,4),(4,2,2),(4,2,4),(4,4,4)]
  do_path<0,0,0>(cg.cgr[0],  cg.cgl[0],  s_in0, s_in0, s_out0, s_X, s_Y, s_dump);
  do_path<0,2,2>(cg.cgr[1],  cg.cgl[1],  s_in2, s_in2, s_out0, s_X, s_Y, s_dump);
  do_path<0,4,4>(cg.cgr[2],  cg.cgl[2],  s_in4, s_in4, s_out0, s_X, s_Y, s_dump);
  do_path<2,0,2>(cg.cgr[3],  cg.cgl[3],  s_in0, s_in2, s_out2, s_X, s_Y, s_dump);
  do_path<2,2,2>(cg.cgr[4],  cg.cgl[4],  s_in2, s_in2, s_out2, s_X, s_Y, s_dump);
  do_path<2,2,4>(cg.cgr[5],  cg.cgl[5],  s_in2, s_in4, s_out2, s_X, s_Y, s_dump);
  do_path<2,4,4>(cg.cgr[6],  cg.cgl[6],  s_in4, s_in4, s_out2, s_X, s_Y, s_dump);
  do_path<4,0,4>(cg.cgr[7],  cg.cgl[7],  s_in0, s_in4, s_out4, s_X, s_Y, s_dump);
  do_path<4,2,2>(cg.cgr[8],  cg.cgl[8],  s_in2, s_in2, s_out4, s_X, s_Y, s_dump);
  do_path<4,2,4>(cg.cgr[9],  cg.cgl[9],  s_in2, s_in4, s_out4, s_X, s_Y, s_dump);
  do_path<4,4,4>(cg.cgr[10], cg.cgl[10], s_in4, s_in4, s_out4, s_X, s_Y, s_dump);

  // outputs: rh_n[0] | rh_n[2] | rh_n[4] | feats, concatenated flat
  float* o0 = out;
  float* o2 = out + NVOX;
  float* o4 = out + NVOX + NVOX*25;
  float* of = out + NVOX + NVOX*25 + NVOX*81;
  {
    float4* d0 = (float4*)(o0 + wg*16);
    float4* d2 = (float4*)(o2 + wg*(16*25));
    float4* d4 = (float4*)(o4 + wg*(16*81));
    const float4* p0 = (const float4*)s_out0;
    const float4* p2 = (const float4*)s_out2;
    const float4* p4 = (const float4*)s_out4;
    for (int t = lane; t < 4;   t += 32) d0[t] = p0[t];
    for (int t = lane; t < 100; t += 32) d2[t] = p2[t];
    for (int t = lane; t < 324; t += 32) d4[t] = p4[t];
  }

  // feats[n, lblk*ABC + abc] = (8*pi^2/(2l+1)) * sum_{k,j} rh_n[l][v,k,j]^2
  if (lane < 16) {
    const int vg  = wg*16 + lane;
    const int n   = vg / ABC;
    const int abc = vg % ABC;
    const float PI8SQ = 78.95683520871486f;   // 8*pi^2
    float s0 = s_out0[lane] * s_out0[lane];
    float s2 = 0.0f;
    #pragma unroll
    for (int t = 0; t < 25; ++t) { float x = s_out2[lane*25 + t]; s2 += x*x; }
    float s4 = 0.0f;
    #pragma unroll
    for (int t = 0; t < 81; ++t) { float x = s_out4[lane*81 + t]; s4 += x*x; }
    of[n*(3*ABC) + 0*ABC + abc] = PI8SQ          * s0;
    of[n*(3*ABC) + 1*ABC + abc] = (PI8SQ / 5.0f) * s2;
    of[n*(3*ABC) + 2*ABC + abc] = (PI8SQ / 9.0f) * s4;
  }
}

extern "C" void kernel_launch(void* const* d_in, const int* in_sizes, int n_in,
                              void* d_out, int out_size, void* d_ws, size_t ws_size,
                              hipStream_t stream) {
  (void)in_sizes; (void)n_in; (void)out_size; (void)d_ws; (void)ws_size;
  const float* rh0 = (const float*)d_in[0];
  const float* rh2 = (const float*)d_in[1];
  const float* rh4 = (const float*)d_in[2];
  CGPtrs cg;
  // setup_inputs() dict order: rh0, rh2, rh4, then (cgr, cgl) interleaved per path
  for (int p = 0; p < 11; ++p) {
    cg.cgr[p] = (const float*)d_in[3 + 2*p];
    cg.cgl[p] = (const float*)d_in[4 + 2*p];
  }
  qnl_kernel<<<TILES, 32, 0, stream>>>(rh0, rh2, rh4, cg, (float*)d_out);
}